// ToyModel_44976897524154
// MI455X (gfx1250) — compile-verified
//
#include <hip/hip_runtime.h>
#include <hip/hip_bf16.h>

typedef __attribute__((ext_vector_type(2))) float v2f;
typedef __attribute__((ext_vector_type(8))) float v8f;

// ---------------------------------------------------------------------------
// Kernel 1: per-image reduction.
// For each image b (64x64x1), compute:
//   T = total sum, R0/R63 = first/last row sums, C0/C63 = first/last col sums
// Then S[b,k] for the 9 conv taps k=(dh,dw):
//   S = T - (dh==0 ? R63 : dh==2 ? R0 : 0) - (dw==0 ? C63 : dw==2 ? C0 : 0) + corner
// Scaled by 1/4096 (the GAP mean) and written to a [512 x 12] f32 A-staging
// matrix (K padded 9 -> 12 with zeros for the 16x16x4 WMMA chain).
// ---------------------------------------------------------------------------
__global__ void __launch_bounds__(256) bnn_reduce_kernel(
    const float* __restrict__ x, float* __restrict__ A) {
  const int b = blockIdx.x;
  const float* __restrict__ img = x + (size_t)b * 4096;
  const int t = threadIdx.x;

  float tot = 0.f, r0 = 0.f, r63 = 0.f, c0 = 0.f, c63 = 0.f;
#pragma unroll
  for (int i = 0; i < 16; ++i) {
    const int e = t + i * 256;           // coalesced: consecutive lanes -> consecutive addrs
    const float v = img[e];
    const int h = e >> 6;
    const int w = e & 63;
    tot += v;
    if (h == 0)  r0  += v;
    if (h == 63) r63 += v;
    if (w == 0)  c0  += v;
    if (w == 63) c63 += v;
  }

  // wave32 shuffle reduction
#pragma unroll
  for (int m = 16; m >= 1; m >>= 1) {
    tot += __shfl_xor(tot, m, 32);
    r0  += __shfl_xor(r0,  m, 32);
    r63 += __shfl_xor(r63, m, 32);
    c0  += __shfl_xor(c0,  m, 32);
    c63 += __shfl_xor(c63, m, 32);
  }

  __shared__ float sm[8][5];
  const int wave = t >> 5;
  if ((t & 31) == 0) {
    sm[wave][0] = tot; sm[wave][1] = r0; sm[wave][2] = r63;
    sm[wave][3] = c0;  sm[wave][4] = c63;
  }
  __syncthreads();

  if (t == 0) {
    float T = 0.f, R0 = 0.f, R63 = 0.f, C0 = 0.f, C63 = 0.f;
#pragma unroll
    for (int i = 0; i < 8; ++i) {
      T   += sm[i][0]; R0 += sm[i][1]; R63 += sm[i][2];
      C0  += sm[i][3]; C63 += sm[i][4];
    }
    // corner pixels
    const float x00 = img[0];          // (0,0)
    const float x0N = img[63];         // (0,63)
    const float xN0 = img[4032];       // (63,0)
    const float xNN = img[4095];       // (63,63)

    const float inv = 1.0f / 4096.0f;
    float* __restrict__ row = A + (size_t)b * 12;
#pragma unroll
    for (int dh = 0; dh < 3; ++dh) {
#pragma unroll
      for (int dw = 0; dw < 3; ++dw) {
        const float a = (dh == 0) ? R63 : (dh == 2) ? R0 : 0.f;
        const float bb = (dw == 0) ? C63 : (dw == 2) ? C0 : 0.f;
        float corner = 0.f;
        if (dh == 0 && dw == 0) corner = xNN;
        if (dh == 0 && dw == 2) corner = xN0;
        if (dh == 2 && dw == 0) corner = x0N;
        if (dh == 2 && dw == 2) corner = x00;
        row[dh * 3 + dw] = (T - a - bb + corner) * inv;
      }
    }
    row[9] = 0.f; row[10] = 0.f; row[11] = 0.f;  // K padding
  }
}

// ---------------------------------------------------------------------------
// Kernel 2: fold sign(conv_w) into dense_w:
//   M[k,n] = sum_c sign(conv_w[k,c]) * dense_w[c,n]    (12x16, zero padded)
//   bias2[n] = sum_c conv_b[c]*dense_w[c,n] + dense_b[n]
// ---------------------------------------------------------------------------
__global__ void __launch_bounds__(256) bnn_prep_kernel(
    const float* __restrict__ conv_w, const float* __restrict__ conv_b,
    const float* __restrict__ dense_w, const float* __restrict__ dense_b,
    float* __restrict__ M, float* __restrict__ bias2) {
  const int t = threadIdx.x;
  if (t < 192) {
    const int k = t >> 4;      // 0..11
    const int n = t & 15;      // 0..15
    float acc = 0.f;
    if (k < 9 && n < 10) {
#pragma unroll
      for (int c = 0; c < 32; ++c) {
        const float w = conv_w[k * 32 + c];        // [3,3,1,32] flat: k*32+c
        const float s = (w >= 0.f) ? 1.0f : -1.0f; // ste_sign: sign(0)=+1
        acc += s * dense_w[c * 10 + n];
      }
    }
    M[k * 16 + n] = acc;
  } else if (t < 208) {
    const int n = t - 192;     // 0..15
    float acc = 0.f;
    if (n < 10) {
#pragma unroll
      for (int c = 0; c < 32; ++c) acc += conv_b[c] * dense_w[c * 10 + n];
      acc += dense_b[n];
    }
    bias2[n] = acc;
  }
}

// ---------------------------------------------------------------------------
// Kernel 3: logits = S @ M via V_WMMA_F32_16X16X4_F32 (3 chained K=4 steps),
// then bias + softmax over the 10 classes, entirely in registers.
// One wave (32 lanes, full EXEC) per 16-row tile; 512/16 = 32 waves total.
//
// Operand layout (ISA 7.12.2, 32-bit, wave32):
//   A 16x4:  lane&15 = M row; v0 = K=(lane<16?0:2), v1 = K=(lane<16?1:3)
//   B 4x16:  lane&15 = N col; v0 = K=(lane<16?0:2), v1 = K=(lane<16?1:3)
//   C/D 16x16: vgpr v, lane l -> M = v + (l>=16 ? 8 : 0), N = l&15
// ---------------------------------------------------------------------------
__global__ void __launch_bounds__(32) bnn_wmma_softmax_kernel(
    const float* __restrict__ A, const float* __restrict__ Bm,
    const float* __restrict__ bias2, float* __restrict__ out) {
  const int lane = threadIdx.x;        // 0..31, one full wave
  const int r = lane & 15;
  const int kh = (lane >> 4) << 1;     // 0 or 2
  const int row0 = blockIdx.x * 16;

  const float* __restrict__ Ar = A + (size_t)(row0 + r) * 12;

  v8f acc = {};
#pragma unroll
  for (int kb = 0; kb < 12; kb += 4) {
    v2f a, bv;
    a.x  = Ar[kb + kh];
    a.y  = Ar[kb + kh + 1];
    bv.x = Bm[(kb + kh) * 16 + r];
    bv.y = Bm[(kb + kh + 1) * 16 + r];
    // 8 args: (neg_a, A, neg_b, B, c_mod, C, reuse_a, reuse_b)
    acc = __builtin_amdgcn_wmma_f32_16x16x4_f32(
        false, a, false, bv, (short)0, acc, false, false);
  }

  const float bias = bias2[r];
  const bool valid = (r < 10);
  const int mbase = row0 + ((lane >> 4) << 3);

#pragma unroll
  for (int v = 0; v < 8; ++v) {
    float xv = valid ? (acc[v] + bias) : -3.0e38f;
    // max over the 16-lane N-group (xor masks < 16 stay inside the half)
    float mx = xv;
#pragma unroll
    for (int m = 8; m >= 1; m >>= 1) mx = fmaxf(mx, __shfl_xor(mx, m, 32));
    float e = valid ? expf(xv - mx) : 0.f;
    float s = e;
#pragma unroll
    for (int m = 8; m >= 1; m >>= 1) s += __shfl_xor(s, m, 32);
    if (valid) {
      out[(size_t)(mbase + v) * 10 + r] = e / s;
    }
  }
}

// ---------------------------------------------------------------------------
extern "C" void kernel_launch(void* const* d_in, const int* in_sizes, int n_in,
                              void* d_out, int out_size, void* d_ws, size_t ws_size,
                              hipStream_t stream) {
  const float* x       = (const float*)d_in[0]; // [512,64,64,1]
  const float* conv_w  = (const float*)d_in[1]; // [3,3,1,32]
  const float* conv_b  = (const float*)d_in[2]; // [32]
  const float* dense_w = (const float*)d_in[3]; // [32,10]
  const float* dense_b = (const float*)d_in[4]; // [10]
  float* out = (float*)d_out;                   // [512,10]

  float* ws    = (float*)d_ws;
  float* A     = ws;                 // 512 x 12 f32 (24 KB)
  float* M     = ws + 512 * 12;      // 12 x 16 f32
  float* bias2 = M + 12 * 16;        // 16 f32

  bnn_reduce_kernel<<<512, 256, 0, stream>>>(x, A);
  bnn_prep_kernel<<<1, 256, 0, stream>>>(conv_w, conv_b, dense_w, dense_b, M, bias2);
  bnn_wmma_softmax_kernel<<<32, 32, 0, stream>>>(A, M, bias2, out);
}